// VQEmbedding_8813272891801
// MI455X (gfx1250) — compile-verified
//
#include <hip/hip_runtime.h>

// ---------------------------------------------------------------------------
// VQ-VAE straight-through forward for MI455X (gfx1250):
//   dist[N,K] = ||x||^2 + ||c||^2 - 2 x.c^T   via V_WMMA_F32_16X16X4_F32
//   logits = -dist, indices = argmin_K, z_q = gather(codebook, indices)
// Codebook tiles streamed into LDS by the Tensor Data Mover (double-buffered,
// synchronized with s_wait_tensorcnt). f32 WMMA keeps reference precision.
// A tile stored r-major (stride 260 dwords) so A-frags are single ds_load_b64,
// conflict-free across all 64 banks; B tile row-padded by the TDM pad feature.
// ---------------------------------------------------------------------------

typedef float v2f  __attribute__((ext_vector_type(2)));
typedef float v8f  __attribute__((ext_vector_type(8)));
typedef unsigned int u32x4 __attribute__((ext_vector_type(4)));
typedef int   i32x4 __attribute__((ext_vector_type(4)));
typedef int   i32x8 __attribute__((ext_vector_type(8)));

#define D_DIM   256
#define K_CODES 8192
#define HW      4096
#define TM      128                 // rows (queries) per block
#define TN      64                  // codes per K-tile
#define NTILES  (K_CODES / TN)      // 128
#define SA      260                 // A_lds stride (dwords): 256 + 4 pad (r-major)
#define SB      260                 // B_lds stride (dwords): 256 + 4 pad (TDM pad)
#define SD      132                 // dist_lds stride (dwords): 128 rows + 4 pad

// Issue one TDM load: 64 codebook rows x 256 f32, row-padded by 4 dwords in LDS.
__device__ __forceinline__ void tdm_load_codebook_tile(const float* gsrc,
                                                       unsigned lds_byte) {
#if __has_builtin(__builtin_amdgcn_tensor_load_to_lds)
  unsigned long long ga = (unsigned long long)(const void*)gsrc;
  u32x4 g0;
  g0[0] = 1u;                                             // count=1, user mode
  g0[1] = lds_byte;                                       // lds_addr
  g0[2] = (unsigned)ga;                                   // global_addr[31:0]
  g0[3] = (unsigned)((ga >> 32) & 0x1FFFFFFull)           // global_addr[56:32]
        | (2u << 30);                                     // type = 2 (image)
  i32x8 g1;
  g1[0] = (int)((2u << 16)      // data_size = 4B
              | (1u << 20)      // pad_enable
              | (7u << 22)      // pad_interval = 256 dwords
              | (3u << 25));    // pad_amount   = 4 dwords
  g1[1] = (int)((unsigned)(D_DIM & 0xFFFF) << 16);        // tensor_dim0 lo
  g1[2] = (int)((D_DIM >> 16) | ((unsigned)(K_CODES & 0xFFFF) << 16)); // dim0 hi | dim1 lo
  g1[3] = (int)((K_CODES >> 16) | ((unsigned)D_DIM << 16));            // dim1 hi | tile_dim0
  g1[4] = TN;                                             // tile_dim1 = 64, tile_dim2 = 0
  g1[5] = D_DIM;                                          // tensor_dim0_stride
  g1[6] = 0;
  g1[7] = 0;
  i32x4 gz = {0, 0, 0, 0};
#if defined(__clang_major__) && __clang_major__ >= 23
  i32x8 gz8 = {0, 0, 0, 0, 0, 0, 0, 0};
  __builtin_amdgcn_tensor_load_to_lds(g0, g1, gz, gz, gz8, 0);
#else
  __builtin_amdgcn_tensor_load_to_lds(g0, g1, gz, gz, 0);
#endif
#else
  (void)gsrc; (void)lds_byte;
#endif
}

__device__ __forceinline__ void tdm_wait0() {
#if __has_builtin(__builtin_amdgcn_s_wait_tensorcnt)
  __builtin_amdgcn_s_wait_tensorcnt(0);
#endif
}

// ---------------------------------------------------------------------------
// Kernel 0: per-code squared norms -> workspace
// ---------------------------------------------------------------------------
__global__ void vq_csq_kernel(const float* __restrict__ cb,
                              float* __restrict__ csq) {
  int k = blockIdx.x * blockDim.x + threadIdx.x;
  const float4* row = (const float4*)(cb + (size_t)k * D_DIM);
  float s = 0.f;
  for (int j = 0; j < D_DIM / 4; ++j) {
    float4 v = row[j];
    s += v.x * v.x + v.y * v.y + v.z * v.z + v.w * v.w;
  }
  csq[k] = s;
}

// ---------------------------------------------------------------------------
// Main fused kernel: WMMA f32 GEMM + logits + argmin + gather
// ---------------------------------------------------------------------------
__global__ __launch_bounds__(256, 1)
void vq_wmma_kernel(const float* __restrict__ z,
                    const float* __restrict__ cb,
                    const float* __restrict__ csq,
                    float* __restrict__ zq,
                    float* __restrict__ zqb,
                    float* __restrict__ logits) {
  __shared__ float A_lds[TM * SA];           // 133120 B, r-major x-tile
  __shared__ float B_lds[2][TN * SB];        // 2 x 66560 B, code-major (TDM)
  __shared__ float dist_lds[TN * SD];        // 33792 B, col-major dist tile
  __shared__ float xsq_lds[TM];
  __shared__ int   idx_lds[TM];

  const int tid  = threadIdx.x;
  const int lane = tid & 31;
  const int wid  = tid >> 5;
  const int half = lane >> 4;                // 0: k+0/1, 1: k+2/3 (f32 WMMA frag)
  const int l    = lane & 15;
  const int wm   = wid >> 1;                 // 0..3 : 32-row stripe
  const int wn   = wid & 1;                  // 0..1 : 32-col stripe

  const int n0  = blockIdx.x * TM;           // first flat (b,h,w) row
  const int b   = n0 >> 12;                  // n0 / HW
  const int rem = n0 & (HW - 1);
  const float* zblk = z + (size_t)b * (D_DIM * HW) + rem;

  // ---- stage A tile (x vectors) into LDS, r-major ----
  // Global reads: coalesced float4 runs along hw; LDS: 4 scalar row-scatters
  // (one-time; bank conflicts here are irrelevant vs. 262144 WMMAs/block).
  for (int i = tid; i < D_DIM * (TM / 4); i += 256) {
    int d  = i >> 5;
    int r4 = (i & 31) << 2;
    float4 v = *(const float4*)(zblk + (size_t)d * HW + r4);
    A_lds[(r4 + 0) * SA + d] = v.x;
    A_lds[(r4 + 1) * SA + d] = v.y;
    A_lds[(r4 + 2) * SA + d] = v.z;
    A_lds[(r4 + 3) * SA + d] = v.w;
  }
  __syncthreads();

  // ---- per-row ||x||^2 ----
  if (tid < TM) {
    float s = 0.f;
    for (int d = 0; d < D_DIM; ++d) {
      float a = A_lds[tid * SA + d];
      s += a * a;
    }
    xsq_lds[tid] = s;
  }

  float best = 3.4e38f;
  int   bidx = 0;

  // ---- prologue: async-load first codebook tile ----
  if (wid == 0) {
    tdm_load_codebook_tile(cb, (unsigned)(size_t)(void*)&B_lds[0][0]);
  }
  __syncthreads();

  const int mA0 = wm * 32 + l;
  const int mA1 = mA0 + 16;
  const int cB0 = wn * 32 + l;
  const int cB1 = cB0 + 16;

  for (int kt = 0; kt < NTILES; ++kt) {
    const int k0 = kt * TN;
    if (wid == 0) tdm_wait0();               // tile kt landed in LDS
    __syncthreads();                         // publish; fences prior epilogue
    if (wid == 0 && (kt + 1) < NTILES) {     // prefetch tile kt+1 (overlaps)
      tdm_load_codebook_tile(cb + (size_t)(k0 + TN) * D_DIM,
                             (unsigned)(size_t)(void*)&B_lds[(kt + 1) & 1][0]);
    }
    const float* Bt = B_lds[kt & 1];

    // hoisted: code norms for this wave's columns (latency hides under WMMA)
    const float cs0 = csq[k0 + cB0];
    const float cs1 = csq[k0 + cB1];

    v8f acc00 = {0, 0, 0, 0, 0, 0, 0, 0};
    v8f acc01 = acc00, acc10 = acc00, acc11 = acc00;

    for (int kk = 0; kk < D_DIM; kk += 4) {
      const int ka = kk + half * 2;
      v2f a0 = *(const v2f*)&A_lds[mA0 * SA + ka];
      v2f a1 = *(const v2f*)&A_lds[mA1 * SA + ka];
      v2f b0 = *(const v2f*)&Bt[cB0 * SB + ka];
      v2f b1 = *(const v2f*)&Bt[cB1 * SB + ka];
      acc00 = __builtin_amdgcn_wmma_f32_16x16x4_f32(false, a0, false, b0,
                                                    (short)0, acc00, false, false);
      acc01 = __builtin_amdgcn_wmma_f32_16x16x4_f32(false, a0, false, b1,
                                                    (short)0, acc01, false, false);
      acc10 = __builtin_amdgcn_wmma_f32_16x16x4_f32(false, a1, false, b0,
                                                    (short)0, acc10, false, false);
      acc11 = __builtin_amdgcn_wmma_f32_16x16x4_f32(false, a1, false, b1,
                                                    (short)0, acc11, false, false);
    }

    // ---- dist = ||x||^2 + ||c||^2 - 2 acc  -> LDS (column-major) ----
    const int row0 = wm * 32 + half * 8;     // lanes 0-15: rows v, 16-31: v+8
    {
      v8f a;
      float4 d0, d1;
      a = acc00;
      d0.x = xsq_lds[row0 + 0] + cs0 - 2.f * a[0];
      d0.y = xsq_lds[row0 + 1] + cs0 - 2.f * a[1];
      d0.z = xsq_lds[row0 + 2] + cs0 - 2.f * a[2];
      d0.w = xsq_lds[row0 + 3] + cs0 - 2.f * a[3];
      d1.x = xsq_lds[row0 + 4] + cs0 - 2.f * a[4];
      d1.y = xsq_lds[row0 + 5] + cs0 - 2.f * a[5];
      d1.z = xsq_lds[row0 + 6] + cs0 - 2.f * a[6];
      d1.w = xsq_lds[row0 + 7] + cs0 - 2.f * a[7];
      *(float4*)&dist_lds[cB0 * SD + row0]     = d0;
      *(float4*)&dist_lds[cB0 * SD + row0 + 4] = d1;
      a = acc01;
      d0.x = xsq_lds[row0 + 0] + cs1 - 2.f * a[0];
      d0.y = xsq_lds[row0 + 1] + cs1 - 2.f * a[1];
      d0.z = xsq_lds[row0 + 2] + cs1 - 2.f * a[2];
      d0.w = xsq_lds[row0 + 3] + cs1 - 2.f * a[3];
      d1.x = xsq_lds[row0 + 4] + cs1 - 2.f * a[4];
      d1.y = xsq_lds[row0 + 5] + cs1 - 2.f * a[5];
      d1.z = xsq_lds[row0 + 6] + cs1 - 2.f * a[6];
      d1.w = xsq_lds[row0 + 7] + cs1 - 2.f * a[7];
      *(float4*)&dist_lds[cB1 * SD + row0]     = d0;
      *(float4*)&dist_lds[cB1 * SD + row0 + 4] = d1;
      a = acc10;
      d0.x = xsq_lds[row0 + 16] + cs0 - 2.f * a[0];
      d0.y = xsq_lds[row0 + 17] + cs0 - 2.f * a[1];
      d0.z = xsq_lds[row0 + 18] + cs0 - 2.f * a[2];
      d0.w = xsq_lds[row0 + 19] + cs0 - 2.f * a[3];
      d1.x = xsq_lds[row0 + 20] + cs0 - 2.f * a[4];
      d1.y = xsq_lds[row0 + 21] + cs0 - 2.f * a[5];
      d1.z = xsq_lds[row0 + 22] + cs0 - 2.f * a[6];
      d1.w = xsq_lds[row0 + 23] + cs0 - 2.f * a[7];
      *(float4*)&dist_lds[cB0 * SD + row0 + 16] = d0;
      *(float4*)&dist_lds[cB0 * SD + row0 + 20] = d1;
      a = acc11;
      d0.x = xsq_lds[row0 + 16] + cs1 - 2.f * a[0];
      d0.y = xsq_lds[row0 + 17] + cs1 - 2.f * a[1];
      d0.z = xsq_lds[row0 + 18] + cs1 - 2.f * a[2];
      d0.w = xsq_lds[row0 + 19] + cs1 - 2.f * a[3];
      d1.x = xsq_lds[row0 + 20] + cs1 - 2.f * a[4];
      d1.y = xsq_lds[row0 + 21] + cs1 - 2.f * a[5];
      d1.z = xsq_lds[row0 + 22] + cs1 - 2.f * a[6];
      d1.w = xsq_lds[row0 + 23] + cs1 - 2.f * a[7];
      *(float4*)&dist_lds[cB1 * SD + row0 + 16] = d0;
      *(float4*)&dist_lds[cB1 * SD + row0 + 20] = d1;
    }
    __syncthreads();

    // waves 0-3: running argmin per row; waves 4-7: stream logits = -dist
    if (tid < TM) {
      for (int c = 0; c < TN; ++c) {
        float dv = dist_lds[c * SD + tid];
        if (dv < best) { best = dv; bidx = k0 + c; }
      }
    } else {
      const int t = tid - TM;
      float* lg = logits + (size_t)b * ((size_t)K_CODES * HW) + rem;
      for (int it = 0; it < 16; ++it) {
        int idx = it * 128 + t;
        int c   = idx >> 5;
        int r4  = (idx & 31) << 2;
        float4 v = *(const float4*)&dist_lds[c * SD + r4];
        float4 o;
        o.x = -v.x; o.y = -v.y; o.z = -v.z; o.w = -v.w;
        *(float4*)(lg + (size_t)(k0 + c) * HW + r4) = o;
      }
    }
    // top-of-loop barrier separates these reads from next tile's writes
  }

  // ---- gather winning codes into z_q_x and z_q_x_bar ----
  if (tid < TM) idx_lds[tid] = bidx;
  __syncthreads();
  {
    const int t  = tid & 127;
    const int h2 = tid >> 7;
    const int code = idx_lds[t];
    const size_t obase = (size_t)b * (D_DIM * HW) + rem + t;
    for (int dd = h2; dd < D_DIM; dd += 2) {
      float v = cb[(size_t)code * D_DIM + dd];
      zq[obase + (size_t)dd * HW]  = v;
      zqb[obase + (size_t)dd * HW] = v;
    }
  }
}

extern "C" void kernel_launch(void* const* d_in, const int* in_sizes, int n_in,
                              void* d_out, int out_size, void* d_ws, size_t ws_size,
                              hipStream_t stream) {
  (void)in_sizes; (void)n_in; (void)out_size; (void)ws_size;
  const float* z  = (const float*)d_in[0];   // [8,256,64,64]
  const float* cb = (const float*)d_in[1];   // [8192,256]
  float* out    = (float*)d_out;
  float* zq     = out;                        // [8,256,64,64]
  float* zqb    = out + (size_t)2097152;      // [8,256,64,64]
  float* logits = out + (size_t)4194304;      // [8,8192,64,64]
  float* csq    = (float*)d_ws;               // [8192]

  vq_csq_kernel<<<K_CODES / 256, 256, 0, stream>>>(cb, csq);
  vq_wmma_kernel<<<(8 * HW) / TM, 256, 0, stream>>>(z, cb, csq, zq, zqb, logits);
}